// DeeperGCN_84765474554365
// MI455X (gfx1250) — compile-verified
//
#include <hip/hip_runtime.h>
#include <hip/hip_bf16.h>
#include <math.h>

#define NNODES   100000
#define NEDGES   600000
#define FDIM     256
#define HDIM     128
#define NCLS     47
#define NCLSPAD  64
#define NSEEDS   10000
#define GEN_EPS  1e-7f
#define LN_EPS   1e-5f

typedef float v2f __attribute__((ext_vector_type(2)));
typedef float v8f __attribute__((ext_vector_type(8)));

// ---------------------------------------------------------------------------
// fp32 WMMA GEMM: out[M,Nc] = A[M,K] @ W[K,Nc] + bias[Nc] (+ res[M,Nc])
// Requirements (guaranteed by caller): M % 16 == 0, Nc % 64 == 0, K % 4 == 0,
// Wt is the K x Nc weight matrix stored TRANSPOSED as [Nc][K].
// One wave computes a 16x64 output strip (4 WMMA tiles) -> A fragment reused 4x.
// blockDim = (32,4): 4 waves = 4 row tiles; grid.y indexes 64-column strips.
// A frag (16x4):  lane m=lane&15, a[v] = A[m][k + v + 2*(lane>>4)]
// B frag (4x16):  lane n=lane&15, b[v] = W[k + v + 2*(lane>>4)][n]  (b64 from Wt)
// C/D  (16x16):   lane n=lane&15, d[i] = C[i + 8*(lane>>4)][n]
// ---------------------------------------------------------------------------
__global__ __launch_bounds__(128)
void gemm_f32_wmma_kernel(const float* __restrict__ A, const float* __restrict__ Wt,
                          const float* __restrict__ bias, const float* __restrict__ res,
                          float* __restrict__ out, int M, int K, int Nc)
{
    const int lane  = threadIdx.x;                  // 0..31
    const int mtile = blockIdx.x * 4 + threadIdx.y;
    const int row0  = mtile * 16;
    if (row0 >= M) return;                          // wave-uniform exit
    const int hb   = lane >> 4;                     // 0/1 half-wave
    const int lo   = lane & 15;
    const int arow = row0 + lo;                     // always < M (M % 16 == 0)
    const int col0 = blockIdx.y * 64 + lo;          // tiles at col0 + {0,16,32,48}

    const float* __restrict__ a_ptr = A  + (size_t)arow * K;
    const float* __restrict__ b_ptr = Wt + (size_t)col0 * K;
    const size_t bstep = (size_t)16 * K;

    v8f acc[4];
#pragma unroll
    for (int t = 0; t < 4; ++t)
#pragma unroll
        for (int i = 0; i < 8; ++i) acc[t][i] = 0.f;

#pragma unroll 4
    for (int k = 0; k < K; k += 4) {
        const int ka = k + 2 * hb;
        const v2f a  = *(const v2f*)(a_ptr + ka);
        const v2f b0 = *(const v2f*)(b_ptr + ka);
        const v2f b1 = *(const v2f*)(b_ptr + bstep + ka);
        const v2f b2 = *(const v2f*)(b_ptr + 2 * bstep + ka);
        const v2f b3 = *(const v2f*)(b_ptr + 3 * bstep + ka);
        acc[0] = __builtin_amdgcn_wmma_f32_16x16x4_f32(false, a, false, b0, (short)0, acc[0], false, false);
        acc[1] = __builtin_amdgcn_wmma_f32_16x16x4_f32(false, a, false, b1, (short)0, acc[1], false, false);
        acc[2] = __builtin_amdgcn_wmma_f32_16x16x4_f32(false, a, false, b2, (short)0, acc[2], false, false);
        acc[3] = __builtin_amdgcn_wmma_f32_16x16x4_f32(false, a, false, b3, (short)0, acc[3], false, false);
    }

#pragma unroll
    for (int t = 0; t < 4; ++t) {
        const int   col = col0 + 16 * t;
        const float bv  = bias[col];
#pragma unroll
        for (int i = 0; i < 8; ++i) {
            const int r = row0 + i + 8 * hb;
            float v = acc[t][i] + bv;
            if (res) v += res[(size_t)r * Nc + col];
            out[(size_t)r * Nc + col] = v;
        }
    }
}

// Wt[c][k] = (c < Nc) ? W[k][c] : 0   -- weight transpose with column padding
__global__ __launch_bounds__(256)
void transpose_pad_kernel(const float* __restrict__ W, float* __restrict__ Wt,
                          int K, int Nc, int Ncpad)
{
    const int tid = blockIdx.x * 256 + threadIdx.x;
    if (tid >= Ncpad * K) return;
    const int c = tid / K;
    const int k = tid - c * K;
    Wt[(size_t)c * K + k] = (c < Nc) ? W[(size_t)k * Nc + c] : 0.f;
}

__global__ void pad_bias_kernel(const float* __restrict__ b, float* __restrict__ bp,
                                int Nc, int Ncpad)
{
    const int c = threadIdx.x;
    if (c < Ncpad) bp[c] = (c < Nc) ? b[c] : 0.f;
}

// ---------------------------------------------------------------------------
// Edge pass 1: m = relu(h[src]) + eps ; mx[dst] = max(mx[dst], m)
// All m > 0, so float bits compare like unsigned ints (mx initialized to 0).
// ---------------------------------------------------------------------------
__global__ __launch_bounds__(256)
void edge_max_kernel(const float* __restrict__ hin, const int* __restrict__ src,
                     const int* __restrict__ dst, float* __restrict__ mx)
{
    const int tid = blockIdx.x * 256 + threadIdx.x;
    if (tid >= NEDGES * 32) return;
    const int e = tid >> 5;
    const int f = (tid & 31) * 4;
    const int s = src[e];
    const int d = dst[e];
    const float4 hv = *(const float4*)(hin + (size_t)s * HDIM + f);
    unsigned* mp = (unsigned*)(mx + (size_t)d * HDIM + f);
    atomicMax(mp + 0, __float_as_uint(fmaxf(hv.x, 0.f) + GEN_EPS));
    atomicMax(mp + 1, __float_as_uint(fmaxf(hv.y, 0.f) + GEN_EPS));
    atomicMax(mp + 2, __float_as_uint(fmaxf(hv.z, 0.f) + GEN_EPS));
    atomicMax(mp + 3, __float_as_uint(fmaxf(hv.w, 0.f) + GEN_EPS));
}

// ---------------------------------------------------------------------------
// Edge pass 2: e = exp(m - mx[dst]); denom[dst] += e; numer[dst] += m*e
// (alpha = e/denom is per-dst, so agg = numer/(denom+1e-16) afterwards)
// ---------------------------------------------------------------------------
__global__ __launch_bounds__(256)
void edge_sum_kernel(const float* __restrict__ hin, const int* __restrict__ src,
                     const int* __restrict__ dst, const float* __restrict__ mx,
                     float* __restrict__ denom, float* __restrict__ numer)
{
    const int tid = blockIdx.x * 256 + threadIdx.x;
    if (tid >= NEDGES * 32) return;
    const int e = tid >> 5;
    const int f = (tid & 31) * 4;
    const int s = src[e];
    const int d = dst[e];
    const float4 hv = *(const float4*)(hin + (size_t)s * HDIM + f);
    const float4 mv = *(const float4*)(mx + (size_t)d * HDIM + f);
    const size_t o = (size_t)d * HDIM + f;
    float m, ex;
    m = fmaxf(hv.x, 0.f) + GEN_EPS; ex = expf(m - mv.x);
    atomicAdd(denom + o + 0, ex); atomicAdd(numer + o + 0, m * ex);
    m = fmaxf(hv.y, 0.f) + GEN_EPS; ex = expf(m - mv.y);
    atomicAdd(denom + o + 1, ex); atomicAdd(numer + o + 1, m * ex);
    m = fmaxf(hv.z, 0.f) + GEN_EPS; ex = expf(m - mv.z);
    atomicAdd(denom + o + 2, ex); atomicAdd(numer + o + 2, m * ex);
    m = fmaxf(hv.w, 0.f) + GEN_EPS; ex = expf(m - mv.w);
    atomicAdd(denom + o + 3, ex); atomicAdd(numer + o + 3, m * ex);
}

// t = numer/(denom + 1e-16) + hin   (elementwise, float4)
__global__ __launch_bounds__(256)
void combine_kernel(const float* __restrict__ numer, const float* __restrict__ denom,
                    const float* __restrict__ hin, float* __restrict__ t, int n4)
{
    const int i = blockIdx.x * 256 + threadIdx.x;
    if (i >= n4) return;
    const float4 nu = ((const float4*)numer)[i];
    const float4 de = ((const float4*)denom)[i];
    const float4 hv = ((const float4*)hin)[i];
    float4 r;
    r.x = nu.x / (de.x + 1e-16f) + hv.x;
    r.y = nu.y / (de.y + 1e-16f) + hv.y;
    r.z = nu.z / (de.z + 1e-16f) + hv.z;
    r.w = nu.w / (de.w + 1e-16f) + hv.w;
    ((float4*)t)[i] = r;
}

// out = relu(LayerNorm(in) * g + b) : one wave32 per row, 4 features/lane
__global__ __launch_bounds__(256)
void ln_relu_kernel(const float* __restrict__ in, const float* __restrict__ g,
                    const float* __restrict__ b, float* __restrict__ out, int M)
{
    const int row  = blockIdx.x * 8 + (threadIdx.x >> 5);
    const int lane = threadIdx.x & 31;
    if (row >= M) return;                          // wave-uniform
    const float4 v = *(const float4*)(in + (size_t)row * HDIM + lane * 4);
    float s1 = v.x + v.y + v.z + v.w;
    float s2 = v.x * v.x + v.y * v.y + v.z * v.z + v.w * v.w;
#pragma unroll
    for (int off = 16; off > 0; off >>= 1) {
        s1 += __shfl_xor(s1, off, 32);
        s2 += __shfl_xor(s2, off, 32);
    }
    const float mu  = s1 * (1.f / HDIM);
    const float var = s2 * (1.f / HDIM) - mu * mu;
    const float inv = rsqrtf(var + LN_EPS);
    const float4 gg = *(const float4*)(g + lane * 4);
    const float4 bb = *(const float4*)(b + lane * 4);
    float4 r;
    r.x = fmaxf((v.x - mu) * inv * gg.x + bb.x, 0.f);
    r.y = fmaxf((v.y - mu) * inv * gg.y + bb.y, 0.f);
    r.z = fmaxf((v.z - mu) * inv * gg.z + bb.z, 0.f);
    r.w = fmaxf((v.w - mu) * inv * gg.w + bb.w, 0.f);
    *(float4*)(out + (size_t)row * HDIM + lane * 4) = r;
}

// out[i][:] = in[map[i]][:]  (H=128, float4 per thread)
__global__ __launch_bounds__(256)
void gather_kernel(const float* __restrict__ in, const int* __restrict__ map,
                   float* __restrict__ out, int M)
{
    const int tid = blockIdx.x * 256 + threadIdx.x;
    if (tid >= M * 32) return;
    const int i = tid >> 5;
    const int f = (tid & 31) * 4;
    const int j = map[i];
    *(float4*)(out + (size_t)i * HDIM + f) =
        *(const float4*)(in + (size_t)j * HDIM + f);
}

// row-wise log_softmax over NCLS columns (input stride NCLSPAD): wave32 per row
__global__ __launch_bounds__(256)
void logsoftmax_kernel(const float* __restrict__ in, float* __restrict__ out, int M)
{
    const int row  = blockIdx.x * 8 + (threadIdx.x >> 5);
    const int lane = threadIdx.x & 31;
    if (row >= M) return;
    const float NEG = -3.0e38f;
    const float v0 = (lane      < NCLS) ? in[(size_t)row * NCLSPAD + lane     ] : NEG;
    const float v1 = (lane + 32 < NCLS) ? in[(size_t)row * NCLSPAD + lane + 32] : NEG;
    float mx = fmaxf(v0, v1);
#pragma unroll
    for (int off = 16; off > 0; off >>= 1) mx = fmaxf(mx, __shfl_xor(mx, off, 32));
    float se = ((lane      < NCLS) ? expf(v0 - mx) : 0.f)
             + ((lane + 32 < NCLS) ? expf(v1 - mx) : 0.f);
#pragma unroll
    for (int off = 16; off > 0; off >>= 1) se += __shfl_xor(se, off, 32);
    const float lse = logf(se) + mx;
    if (lane      < NCLS) out[(size_t)row * NCLS + lane     ] = v0 - lse;
    if (lane + 32 < NCLS) out[(size_t)row * NCLS + lane + 32] = v1 - lse;
}

// ---------------------------------------------------------------------------
extern "C" void kernel_launch(void* const* d_in, const int* in_sizes, int n_in,
                              void* d_out, int out_size, void* d_ws, size_t ws_size,
                              hipStream_t stream)
{
    (void)in_sizes; (void)n_in; (void)out_size; (void)ws_size;

    const float* x         = (const float*)d_in[0];
    const int*   src       = (const int*)  d_in[1];   // [2, E]
    const int*   dst       = (const int*)  d_in[2];   // [2, E]
    const int*   node_map  = (const int*)  d_in[3];   // [2, N]
    const int*   final_map = (const int*)  d_in[4];   // [NSEEDS]
    const float* enc_w     = (const float*)d_in[5];   // [FDIM, H]
    const float* enc_b     = (const float*)d_in[6];   // [H]
    const float* gcn_w     = (const float*)d_in[7];   // [3, H, H]
    const float* gcn_b     = (const float*)d_in[8];   // [3, H]
    const float* ln_g      = (const float*)d_in[9];   // [3, H]
    const float* ln_b      = (const float*)d_in[10];  // [3, H]
    const float* pred_w    = (const float*)d_in[11];  // [H, NCLS]
    const float* pred_b    = (const float*)d_in[12];  // [NCLS]
    float* out = (float*)d_out;

    const size_t NH = (size_t)NNODES * HDIM;
    float* B0     = (float*)d_ws;          // h
    float* B1     = B0 + NH;               // h2 = relu(LN(h))
    float* B2     = B0 + 2 * NH;           // mx   -> reused as t = agg + hin
    float* B3     = B0 + 3 * NH;           // denom
    float* B4     = B0 + 4 * NH;           // numer -> reused as conv GEMM output
    float* HG     = B0 + 5 * NH;           // gathered seed rows [NSEEDS, H]
    float* HG2    = HG  + (size_t)NSEEDS * HDIM;
    float* SC     = HG2 + (size_t)NSEEDS * HDIM;            // [NSEEDS, NCLSPAD]
    float* WT_ENC = SC  + (size_t)NSEEDS * NCLSPAD;         // [H, FDIM]
    float* WT_G   = WT_ENC + (size_t)HDIM * FDIM;           // 3 x [H, H]
    float* PWT    = WT_G   + (size_t)3 * HDIM * HDIM;       // [NCLSPAD, H]
    float* PB     = PWT    + (size_t)NCLSPAD * HDIM;        // [NCLSPAD]

    // ---- pre-transpose (and pad) all weights ----------------------------
    transpose_pad_kernel<<<(HDIM * FDIM + 255) / 256, 256, 0, stream>>>(
        enc_w, WT_ENC, FDIM, HDIM, HDIM);
    for (int l = 0; l < 3; ++l)
        transpose_pad_kernel<<<(HDIM * HDIM + 255) / 256, 256, 0, stream>>>(
            gcn_w + (size_t)l * HDIM * HDIM, WT_G + (size_t)l * HDIM * HDIM,
            HDIM, HDIM, HDIM);
    transpose_pad_kernel<<<(NCLSPAD * HDIM + 255) / 256, 256, 0, stream>>>(
        pred_w, PWT, HDIM, NCLS, NCLSPAD);
    pad_bias_kernel<<<1, NCLSPAD, 0, stream>>>(pred_b, PB, NCLS, NCLSPAD);

    const dim3 gblk(32, 4, 1);
    const int  mtN   = NNODES / 16;                        // 6250 row tiles
    const dim3 gridN((mtN + 3) / 4, HDIM / 64, 1);         // N x 128 GEMMs
    const int  egrid = (NEDGES * 32 + 255) / 256;
    const int  cgrid = ((int)(NH / 4) + 255) / 256;
    const int  ngrid = (NNODES * 32 + 255) / 256;
    const int  lgrid = (NNODES + 7) / 8;

    // ---- encoder: h = x @ enc_w + enc_b ---------------------------------
    gemm_f32_wmma_kernel<<<gridN, gblk, 0, stream>>>(
        x, WT_ENC, enc_b, nullptr, B0, NNODES, FDIM, HDIM);

    auto conv = [&](const float* hin, int eblk, const float* wt, const float* bias,
                    const float* res, float* outp) {
        hipMemsetAsync(B2, 0, 3 * NH * sizeof(float), stream);  // mx, denom, numer
        edge_max_kernel<<<egrid, 256, 0, stream>>>(
            hin, src + (size_t)eblk * NEDGES, dst + (size_t)eblk * NEDGES, B2);
        edge_sum_kernel<<<egrid, 256, 0, stream>>>(
            hin, src + (size_t)eblk * NEDGES, dst + (size_t)eblk * NEDGES, B2, B3, B4);
        combine_kernel<<<cgrid, 256, 0, stream>>>(B4, B3, hin, B2, (int)(NH / 4));
        gemm_f32_wmma_kernel<<<gridN, gblk, 0, stream>>>(
            B2, wt, bias, res, outp, NNODES, HDIM, HDIM);
    };

    // ---- layer 0: h = (agg(h) + h) @ w0 + b0 (no residual, no gather) ---
    conv(B0, 0, WT_G, gcn_b, nullptr, B0);

    // ---- layers 1..2 (res+ blocks) --------------------------------------
    for (int l = 1; l < 3; ++l) {
        ln_relu_kernel<<<lgrid, 256, 0, stream>>>(
            B0, ln_g + (size_t)(l - 1) * HDIM, ln_b + (size_t)(l - 1) * HDIM, B1, NNODES);
        conv(B1, l - 1, WT_G + (size_t)l * HDIM * HDIM, gcn_b + (size_t)l * HDIM,
             /*res=*/B0, /*out=*/B4);
        gather_kernel<<<ngrid, 256, 0, stream>>>(
            B4, node_map + (size_t)(l - 1) * NNODES, B0, NNODES);
    }

    // ---- head: gather seeds, LN+ReLU, pred GEMM, log_softmax ------------
    const int sgrid = (NSEEDS * 32 + 255) / 256;
    gather_kernel<<<sgrid, 256, 0, stream>>>(B0, final_map, HG, NSEEDS);
    ln_relu_kernel<<<(NSEEDS + 7) / 8, 256, 0, stream>>>(
        HG, ln_g + 2 * HDIM, ln_b + 2 * HDIM, HG2, NSEEDS);
    const int  mtS = NSEEDS / 16;                           // 625
    const dim3 gridS((mtS + 3) / 4, NCLSPAD / 64, 1);       // 157 x 1
    gemm_f32_wmma_kernel<<<gridS, gblk, 0, stream>>>(
        HG2, PWT, PB, nullptr, SC, NSEEDS, HDIM, NCLSPAD);
    logsoftmax_kernel<<<(NSEEDS + 7) / 8, 256, 0, stream>>>(SC, out, NSEEDS);
}